// LMLLoss_43361989820442
// MI455X (gfx1250) — compile-verified
//
#include <hip/hip_runtime.h>
#include <math.h>

#define N_TOPK   5
#define EPSV     1e-4f
#define N_ITER   20
#define BRANCH   10
#define NB       1024
#define NC       32000
#define TPB      256
#define EPT      (NC / TPB)     // 125 elements per thread
#define NWAVES   (TPB / 32)     // 8 wave32 per block

// ---- CDNA5 async global->LDS path (gfx1250) ---------------------------------
#if defined(__has_builtin)
#  if __has_builtin(__builtin_amdgcn_global_load_async_to_lds_b128) && \
      __has_builtin(__builtin_amdgcn_global_load_async_to_lds_b32)
#    define USE_ASYNC_LDS 1
#  endif
#endif

// Builtin param types per hipcc diagnostic:
//   b128 arg0: 'int __vector(4) __device__ *'  -> as(1) v4i*
//   b32  arg0: '__device__ int *'              -> as(1) int*
typedef int v4i __attribute__((vector_size(16)));
typedef __attribute__((address_space(1))) v4i* gp128_t;
typedef __attribute__((address_space(3))) v4i* lp128_t;
typedef __attribute__((address_space(1))) int* gp32_t;
typedef __attribute__((address_space(3))) int* lp32_t;

__device__ __forceinline__ float fast_rcp(float x) {
  return __builtin_amdgcn_rcpf(x);   // v_rcp_f32
}

// -----------------------------------------------------------------------------
// One block per batch row. Row (125KB) parked in LDS; multisection solve for nu.
// -----------------------------------------------------------------------------
extern "C" __global__ void __launch_bounds__(TPB)
lml_row_kernel(const float* __restrict__ x, const int* __restrict__ y,
               float* __restrict__ rowloss)
{
  extern __shared__ float xs[];            // NC floats = 128000 B (dynamic LDS)
  __shared__ float top[TPB * 6];           // per-thread top-6 lists
  __shared__ float wsum[NWAVES * BRANCH];  // per-wave partial sums
  __shared__ float fsum[BRANCH];           // block-level probe sums

  const int tid = threadIdx.x;
  const int row = blockIdx.x;
  const float* __restrict__ xrow = x + (size_t)row * NC;

  // ---- Stage row into LDS -------------------------------------------------
#ifdef USE_ASYNC_LDS
  {
    const char* g = (const char*)xrow;
    char*       l = (char*)xs;
    // 31 rounds of b128 (256 lanes * 16B = 4096B/round) = 126976 B
    #pragma unroll
    for (int i = 0; i < 31; ++i) {
      const int off = i * 4096 + tid * 16;
      __builtin_amdgcn_global_load_async_to_lds_b128(
          (gp128_t)(g + off), (lp128_t)(l + off), 0, 0);
    }
    // tail: 256 lanes * 4B = 1024 B -> total 128000 B
    {
      const int off = 31 * 4096 + tid * 4;
      __builtin_amdgcn_global_load_async_to_lds_b32(
          (gp32_t)(g + off), (lp32_t)(l + off), 0, 0);
    }
#  if __has_builtin(__builtin_amdgcn_s_wait_asynccnt)
    __builtin_amdgcn_s_wait_asynccnt(0);
#  else
    asm volatile("s_wait_asynccnt 0" ::: "memory");
#  endif
  }
#else
  for (int i = 0; i < EPT; ++i) {
    const int c = tid + (i << 8);
    xs[c] = xrow[c];
  }
#endif
  __syncthreads();

  // ---- Exact top-6 per row (bracket for nu) -------------------------------
  float t6[6];
  #pragma unroll
  for (int k = 0; k < 6; ++k) t6[k] = -__builtin_inff();
  for (int i = 0; i < EPT; ++i) {
    const float v = xs[tid + (i << 8)];
    if (v > t6[5]) {
      t6[5] = v;
      #pragma unroll
      for (int k = 5; k > 0; --k) {
        if (t6[k] > t6[k - 1]) { float tmp = t6[k]; t6[k] = t6[k - 1]; t6[k - 1] = tmp; }
      }
    }
  }
  #pragma unroll
  for (int k = 0; k < 6; ++k) top[tid * 6 + k] = t6[k];
  __syncthreads();

  // Tree-merge 256 sorted-descending 6-lists down to one
  for (int s = TPB / 2; s >= 1; s >>= 1) {
    if (tid < s) {
      float a[6], b[6], o[6];
      #pragma unroll
      for (int k = 0; k < 6; ++k) { a[k] = top[tid * 6 + k]; b[k] = top[(tid + s) * 6 + k]; }
      int ia = 0, ib = 0;
      #pragma unroll
      for (int k = 0; k < 6; ++k) {   // ia+ib == k <= 5, both indices stay in range
        const float av = a[ia], bv = b[ib];
        if (av >= bv) { o[k] = av; ++ia; } else { o[k] = bv; ++ib; }
      }
      #pragma unroll
      for (int k = 0; k < 6; ++k) top[tid * 6 + k] = o[k];
    }
    __syncthreads();
  }

  float lo = -top[4] - 7.0f;   // -x_sorted[k-1] - 7
  float hi = -top[5] + 7.0f;   // -x_sorted[k]   + 7

  // ---- Bracketed multisection: solve sum(sigmoid(x+nu)) == N_TOPK ---------
  const float lsden = 1.0f / 9.0f;       // linspace(0,1,10) step
  const int lane = tid & 31, wv = tid >> 5;

  for (int it = 0; it < N_ITER; ++it) {
    const float r = hi - lo;
    if (!(r > EPSV)) break;

    float nus[BRANCH], enu[BRANCH];
    #pragma unroll
    for (int j = 0; j < BRANCH; ++j) {
      nus[j] = lo + r * ((float)j * lsden);
      enu[j] = __expf(-nus[j]);
    }

    float acc[BRANCH];
    #pragma unroll
    for (int j = 0; j < BRANCH; ++j) acc[j] = 0.0f;

    // sigmoid(x+nu_j) = 1 / (1 + e^{-x} * e^{-nu_j}) : 1 exp + 10 rcp per elem
    #pragma unroll 5
    for (int i = 0; i < EPT; ++i) {
      const float ex = __expf(-xs[tid + (i << 8)]);
      #pragma unroll
      for (int j = 0; j < BRANCH; ++j)
        acc[j] += fast_rcp(fmaf(ex, enu[j], 1.0f));
    }

    // wave32 reduction, then fixed-order cross-wave sum (deterministic)
    #pragma unroll
    for (int j = 0; j < BRANCH; ++j) {
      float v = acc[j];
      #pragma unroll
      for (int m = 16; m >= 1; m >>= 1) v += __shfl_xor(v, m, 32);
      if (lane == 0) wsum[wv * BRANCH + j] = v;
    }
    __syncthreads();
    if (tid < BRANCH) {
      float s = 0.0f;
      #pragma unroll
      for (int w = 0; w < NWAVES; ++w) s += wsum[w * BRANCH + tid];
      fsum[tid] = s;
    }
    __syncthreads();

    int cnt = 0;
    #pragma unroll
    for (int j = 0; j < BRANCH; ++j) cnt += (fsum[j] < (float)N_TOPK) ? 1 : 0;
    int il = cnt - 1; if (il < 0) il = 0;
    int ih = il + 1;  if (ih > BRANCH - 1) ih = BRANCH - 1;
    // recompute nus[il]/nus[ih] arithmetically (avoids dynamic reg indexing)
    const float newlo = lo + r * ((float)il * lsden);
    const float newhi = lo + r * ((float)ih * lsden);
    lo = newlo; hi = newhi;
    // fsum is re-written only after next iteration's barrier -> no extra sync
  }

  if (tid == 0) {
    const int cls = y[row];
    const float t = xs[cls] + lo;
    rowloss[row] = log1pf(__expf(-t));   // -log(sigmoid(t))
  }
}

// -----------------------------------------------------------------------------
// Mean of 1024 per-row losses (deterministic tree reduction)
// -----------------------------------------------------------------------------
extern "C" __global__ void __launch_bounds__(TPB)
lml_reduce_kernel(const float* __restrict__ rowloss, float* __restrict__ out)
{
  __shared__ float s[TPB];
  const int t = threadIdx.x;
  float v = rowloss[t] + rowloss[t + 256] + rowloss[t + 512] + rowloss[t + 768];
  s[t] = v;
  __syncthreads();
  for (int st = TPB / 2; st >= 1; st >>= 1) {
    if (t < st) s[t] += s[t + st];
    __syncthreads();
  }
  if (t == 0) out[0] = s[0] * (1.0f / (float)NB);
}

extern "C" void kernel_launch(void* const* d_in, const int* in_sizes, int n_in,
                              void* d_out, int out_size, void* d_ws, size_t ws_size,
                              hipStream_t stream) {
  (void)in_sizes; (void)n_in; (void)out_size; (void)ws_size;
  const float* x = (const float*)d_in[0];
  const int*   y = (const int*)d_in[1];
  float* rowloss = (float*)d_ws;          // 1024 floats of scratch
  float* out     = (float*)d_out;

  lml_row_kernel<<<dim3(NB), dim3(TPB), NC * sizeof(float), stream>>>(x, y, rowloss);
  lml_reduce_kernel<<<dim3(1), dim3(TPB), 0, stream>>>(rowloss, out);
}